// MyAttention_6459630813993
// MI455X (gfx1250) — compile-verified
//
#include <hip/hip_runtime.h>

// fp32 WMMA fragment types (CDNA5 / gfx1250, wave32)
typedef __attribute__((ext_vector_type(2))) float v2f;   // A/B frag of 16x16x4 f32 WMMA
typedef __attribute__((ext_vector_type(8))) float v8f;   // 16x16 f32 C/D tile

// ---------------------------------------------------------------------------
// Generic batched GEMM: C = A * op(B), fp32, via V_WMMA_F32_16X16X4_F32.
// One wave computes a 64x64 output tile (4x4 grid of 16x16 WMMA tiles).
// Used for the Q/K/V projections and the output projection (NN form).
// ---------------------------------------------------------------------------
template <bool TRANSB>
__global__ __launch_bounds__(32)
void wmma_gemm_f32(const float* __restrict__ Abase,
                   const float* __restrict__ Bbase,
                   float* __restrict__ Cbase,
                   int K,
                   long long lda, long long ldb, long long ldc,
                   int zInner,
                   long long sAo, long long sAi,
                   long long sBo, long long sBi,
                   long long sCo, long long sCi)
{
    const int lane = threadIdx.x;        // 0..31 (wave32)
    const int half = lane >> 4;          // selects K pair {0,1} vs {2,3}
    const int l    = lane & 15;          // row (A) / col (B,C) within 16

    const int z  = blockIdx.z;
    const int zo = z / zInner;
    const int zi = z - zo * zInner;

    const float* A = Abase + (long long)zo * sAo + (long long)zi * sAi;
    const float* B = Bbase + (long long)zo * sBo + (long long)zi * sBi;
    float*       C = Cbase + (long long)zo * sCo + (long long)zi * sCi;

    const long long m0 = (long long)blockIdx.x * 64;
    const long long n0 = (long long)blockIdx.y * 64;

    v8f acc[4][4] = {};

    const float* aptr[4];
#pragma unroll
    for (int i = 0; i < 4; ++i)
        aptr[i] = A + (m0 + 16 * i + l) * lda + 2 * half;

    const float* bptr[4];
#pragma unroll
    for (int j = 0; j < 4; ++j) {
        if (TRANSB) {
            bptr[j] = B + (n0 + 16 * j + l) * ldb + 2 * half;
        } else {
            bptr[j] = B + 2 * half * ldb + (n0 + 16 * j + l);
        }
    }

    for (int k0 = 0; k0 < K; k0 += 4) {
        v2f af[4], bf[4];
#pragma unroll
        for (int i = 0; i < 4; ++i)
            af[i] = *(const v2f*)(aptr[i] + k0);

        if (TRANSB) {
#pragma unroll
            for (int j = 0; j < 4; ++j)
                bf[j] = *(const v2f*)(bptr[j] + k0);
        } else {
#pragma unroll
            for (int j = 0; j < 4; ++j) {
                const float* p = bptr[j] + (long long)k0 * ldb;
                v2f t;
                t.x = p[0];
                t.y = p[ldb];
                bf[j] = t;
            }
        }

#pragma unroll
        for (int i = 0; i < 4; ++i)
#pragma unroll
            for (int j = 0; j < 4; ++j)
                acc[i][j] = __builtin_amdgcn_wmma_f32_16x16x4_f32(
                    false, af[i], false, bf[j], (short)0, acc[i][j], false, false);
    }

#pragma unroll
    for (int i = 0; i < 4; ++i)
#pragma unroll
        for (int j = 0; j < 4; ++j) {
            float* crow = C + (m0 + 16 * i + 8 * half) * ldc + (n0 + 16 * j + l);
#pragma unroll
            for (int r = 0; r < 8; ++r)
                crow[(long long)r * ldc] = acc[i][j][r];
        }
}

// ---------------------------------------------------------------------------
// Fused attention core: for one (b,a) head and one 64-row block of n:
//   S[n, m]  = sum_d Q[n,d] * K[m,d]       (streamed to attn, nontemporal)
//   E[n, d] += sum_m S[n,m] * V[m,d]       (accumulated in registers)
// Avoids re-reading the 537MB attn tensor from HBM (the dominant traffic).
// One wave per block. S strips are turned from C/D layout into A-fragment
// layout through a small padded LDS tile (cross-lane transpose).
// K and V fragments for each strip are preloaded up-front so many loads are
// in flight at once (staggered s_wait_loadcnt instead of load->wait->wmma).
// ---------------------------------------------------------------------------
__global__ __launch_bounds__(32)
void attn_fused_f32(const float* __restrict__ Q,    // [b, n, a*64+d], ld 1024
                    const float* __restrict__ Km,   // [b, m, a*64+d], ld 1024
                    const float* __restrict__ V,    // [b, m, a*64+d], ld 1024
                    float* __restrict__ attn,       // [b, a, n, m]
                    float* __restrict__ E)          // [b, n, a*64+d], ld 1024
{
    const int lane = threadIdx.x;
    const int half = lane >> 4;
    const int l    = lane & 15;

    const int z = blockIdx.z;            // b*16 + a
    const int b = z >> 4;
    const int a = z & 15;
    const long long S1 = 1024LL * 1024;  // per-batch merged-matrix elements

    const float* Qb = Q  + (long long)b * S1 + a * 64;
    const float* Kb = Km + (long long)b * S1 + a * 64;
    const float* Vb = V  + (long long)b * S1 + a * 64;
    float* attnb    = attn + (long long)z * S1;
    float* Eb       = E  + (long long)b * S1 + a * 64;

    const int n0 = blockIdx.x * 64;

    // LDS staging tile: 4 sub-tiles of 16x16, row stride padded to 20 dwords
    // (even => 8B-aligned b64 reads; 20 => conflict-free banks for both the
    // C-layout stores and the A-fragment loads).
    __shared__ float sT[4 * 320];

    // Cache Q A-fragments for the whole 64-row block: qf[i][kd]
    // lane (half,l): Q[n0+16i+l][4kd+2half], [..+1]
    v2f qf[4][16];
#pragma unroll
    for (int i = 0; i < 4; ++i) {
        const float* qrow = Qb + (long long)(n0 + 16 * i + l) * 1024 + 2 * half;
#pragma unroll
        for (int kd = 0; kd < 16; ++kd)
            qf[i][kd] = *(const v2f*)(qrow + 4 * kd);
    }

    v8f eacc[4][4] = {};   // E accumulators: rows n (4x16), cols d (4x16)

#pragma unroll 1
    for (int m0 = 0; m0 < 1024; m0 += 16) {
        // ---- preload K fragments for this strip: 16 b64 loads in flight
        v2f kf[16];
        const float* krow = Kb + (long long)(m0 + l) * 1024 + 2 * half;
#pragma unroll
        for (int kd = 0; kd < 16; ++kd)
            kf[kd] = *(const v2f*)(krow + 4 * kd);

        // ---- preload V fragments (consumed only after the LDS turn-around,
        // so their latency overlaps the whole S computation + attn stores)
        v2f vf[4][4];
#pragma unroll
        for (int km = 0; km < 4; ++km) {
            const float* vrow = Vb + (long long)(m0 + 4 * km + 2 * half) * 1024 + l;
#pragma unroll
            for (int j = 0; j < 4; ++j) {
                v2f t;
                t.x = vrow[16 * j];
                t.y = vrow[16 * j + 1024];
                vf[km][j] = t;
            }
        }

        // ---- S strip: rows n0..n0+63, cols m0..m0+15  (NT gemm, K = 64)
        v8f sacc[4] = {};
#pragma unroll
        for (int kd = 0; kd < 16; ++kd) {
#pragma unroll
            for (int i = 0; i < 4; ++i)
                sacc[i] = __builtin_amdgcn_wmma_f32_16x16x4_f32(
                    false, qf[i][kd], false, kf[kd], (short)0, sacc[i], false, false);
        }

        // ---- stream S to attn (write-once => nontemporal) + stage into LDS
        asm volatile("" ::: "memory");
#pragma unroll
        for (int i = 0; i < 4; ++i) {
            float* arow = attnb + (long long)(n0 + 16 * i + 8 * half) * 1024 + m0 + l;
            float* srow = sT + i * 320 + (8 * half) * 20 + l;
#pragma unroll
            for (int r = 0; r < 8; ++r) {
                __builtin_nontemporal_store(sacc[i][r], arow + (long long)r * 1024);
                srow[r * 20] = sacc[i][r];
            }
        }
        // Same-wave LDS ops are processed in order; the explicit wait + fences
        // keep the compiler from reordering the cross-lane turn-around.
        asm volatile("s_wait_dscnt 0x0" ::: "memory");

        // ---- E += S * V : k = m (16 wide -> 4 WMMA k-steps)
#pragma unroll
        for (int km = 0; km < 4; ++km) {
            v2f sa[4];
#pragma unroll
            for (int i = 0; i < 4; ++i)
                sa[i] = *(const v2f*)(sT + i * 320 + l * 20 + 4 * km + 2 * half);

#pragma unroll
            for (int j = 0; j < 4; ++j)
#pragma unroll
                for (int i = 0; i < 4; ++i)
                    eacc[i][j] = __builtin_amdgcn_wmma_f32_16x16x4_f32(
                        false, sa[i], false, vf[km][j], (short)0, eacc[i][j], false, false);
        }
        asm volatile("" ::: "memory");
    }

    // ---- write E in merged-head layout [b, n, a*64+d]
#pragma unroll
    for (int i = 0; i < 4; ++i)
#pragma unroll
        for (int j = 0; j < 4; ++j) {
            float* erow = Eb + (long long)(n0 + 16 * i + 8 * half) * 1024 + 16 * j + l;
#pragma unroll
            for (int r = 0; r < 8; ++r)
                erow[(long long)r * 1024] = eacc[i][j][r];
        }
}

// Problem constants: B=8, N=1024, D_MODEL=1024, A=16, D_K=D_V=64, D_OUT=1024.
extern "C" void kernel_launch(void* const* d_in, const int* in_sizes, int n_in,
                              void* d_out, int out_size, void* d_ws, size_t ws_size,
                              hipStream_t stream)
{
    const float* query = (const float*)d_in[0];   // [8,1024,1024]
    const float* key   = (const float*)d_in[1];
    const float* value = (const float*)d_in[2];
    const float* W_q   = (const float*)d_in[3];   // [1024,16,64] == [1024,1024]
    const float* W_k   = (const float*)d_in[4];
    const float* W_v   = (const float*)d_in[5];
    const float* W_o   = (const float*)d_in[6];   // [1024,1024]

    const long long S = 1024LL * 1024;            // per-batch matrix elements
    const long long T = 8LL * S;                  // per-tensor elements

    float* Qws = (float*)d_ws;                    // [b, n, a*64+d]
    float* Kws = Qws + T;
    float* Vws = Kws + T;
    float* Ews = Vws + T;                         // merged-head attn*V

    float* out_p  = (float*)d_out;                // [8,1024,1024]
    float* attn_p = out_p + T;                    // [8,16,1024,1024]

    dim3 blk(32);

    // Stage 1: Q/K/V projections (NN), per batch [1024x1024] x [1024x1024].
    wmma_gemm_f32<false><<<dim3(16, 16, 8), blk, 0, stream>>>(
        query, W_q, Qws, 1024, 1024, 1024, 1024, 1, S, 0, 0, 0, S, 0);
    wmma_gemm_f32<false><<<dim3(16, 16, 8), blk, 0, stream>>>(
        key, W_k, Kws, 1024, 1024, 1024, 1024, 1, S, 0, 0, 0, S, 0);
    wmma_gemm_f32<false><<<dim3(16, 16, 8), blk, 0, stream>>>(
        value, W_v, Vws, 1024, 1024, 1024, 1024, 1, S, 0, 0, 0, S, 0);

    // Stage 2+3 fused: attn = Q K^T (streamed out) and E = attn V (in regs).
    attn_fused_f32<<<dim3(16, 1, 128), blk, 0, stream>>>(
        Qws, Kws, Vws, attn_p, Ews);

    // Stage 4: out = E * W_o (NN).
    wmma_gemm_f32<false><<<dim3(16, 16, 8), blk, 0, stream>>>(
        Ews, W_o, out_p, 1024, 1024, 1024, 1024, 1, S, 0, 0, 0, S, 0);
}